// GCNII_927712936102
// MI455X (gfx1250) — compile-verified
//
#include <hip/hip_runtime.h>
#include <math.h>

typedef __attribute__((ext_vector_type(2))) float v2f;
typedef __attribute__((ext_vector_type(8))) float v8f;

#define N_NODES 50000
#define N_EDGES 800000
#define NFEAT   500
#define NCLASS  64
#define NHIDDEN 128
#define NLAYERS 16
#define ALPHA   0.1f
#define LAMDA   0.5f

// LDS B-fragment swizzle: for k-group g (4 K-rows), half p=(k>>1)&1, sub s=k&1,
// column n:  idx = ((g*2+p)*N + n)*2 + s.
// A lane fetches its v2f B fragment with ONE aligned ds_load_b64 (pairs of
// fragments combine into ds_load_2addr_b64).
__device__ __forceinline__ int bswz(int k, int n, int N) {
    return (((k >> 2) * 2 + ((k >> 1) & 1)) * N + n) * 2 + (k & 1);
}

#define WMMA_F32(a, b, c) __builtin_amdgcn_wmma_f32_16x16x4_f32( \
    false, (a), false, (b), (short)0, (c), false, false)

// ---------------------------------------------------------------------------
// GEMM1: h0 = relu(x @ w1 + b1)      [50000 x 500] @ [500 x 128]
// 256 thr = 8 waves; block tile 128 x 128; w1 staged in swizzled K-chunks of 100.
// A fragment pipelined one k-step ahead; B fragments batch-staged per step.
// ---------------------------------------------------------------------------
__global__ __launch_bounds__(256) void gemm_in_kernel(
    const float* __restrict__ x, const float* __restrict__ w1,
    const float* __restrict__ b1, float* __restrict__ h0)
{
    __shared__ __align__(16) float lw[100 * NHIDDEN];   // 51.2 KB
    const int tid  = threadIdx.x;
    const int wave = tid >> 5;
    const int lane = tid & 31;
    const int lh   = lane >> 4;
    const int ln   = lane & 15;
    const int rowBase = blockIdx.x * 128 + wave * 16;
    const int row  = rowBase + ln;
    const int rowc = row < N_NODES ? row : N_NODES - 1;   // clamp: keep EXEC full for WMMA
    const float* arow = x + (size_t)rowc * NFEAT + 2 * lh;

    v8f acc[8];
    #pragma unroll
    for (int t = 0; t < 8; ++t)
        #pragma unroll
        for (int i = 0; i < 8; ++i) acc[t][i] = 0.0f;

    for (int kc = 0; kc < NFEAT; kc += 100) {
        __syncthreads();
        for (int i = tid; i < 100 * NHIDDEN; i += 256) {
            int k = i >> 7, n = i & 127;
            lw[bswz(k, n, NHIDDEN)] = w1[(size_t)(kc + k) * NHIDDEN + n];
        }
        __syncthreads();
        v2f a; a.x = arow[kc]; a.y = arow[kc + 1];        // prologue A (g=0)
        #pragma unroll 5
        for (int g = 0; g < 25; ++g) {
            // stage all 8 B fragments (4 x ds_load_2addr_b64, independent regs)
            const v2f* brow = (const v2f*)(lw + (size_t)(g * 2 + lh) * NHIDDEN * 2);
            v2f b[8];
            #pragma unroll
            for (int t = 0; t < 8; ++t) b[t] = brow[t * 16 + ln];
            // pipeline next A fragment over this step's WMMAs
            v2f an;
            if (g < 24) { an.x = arow[kc + 4 * g + 4]; an.y = arow[kc + 4 * g + 5]; }
            __builtin_prefetch(arow + kc + 4 * g + 100, 0, 1);   // next K-chunk
            #pragma unroll
            for (int t = 0; t < 8; ++t) acc[t] = WMMA_F32(a, b[t], acc[t]);
            a = an;
        }
    }
    #pragma unroll
    for (int t = 0; t < 8; ++t) {
        #pragma unroll
        for (int r = 0; r < 8; ++r) {
            int orow = rowBase + r + 8 * lh;      // C/D layout: VGPR r -> M=r / r+8
            int ocol = t * 16 + ln;
            if (orow < N_NODES) {
                float v = acc[t][r] + b1[ocol];
                h0[(size_t)orow * NHIDDEN + ocol] = v > 0.0f ? v : 0.0f;
            }
        }
    }
}

// ---------------------------------------------------------------------------
// SpMM: hi[dst] += w_e * h[src].  One wave/edge; 32 lanes x float4 = 128 feats.
// h, hi are L2-resident (25.6 MB each vs 192 MB L2) -> runs at L2 bandwidth.
// ---------------------------------------------------------------------------
__global__ __launch_bounds__(256) void spmm_kernel(
    const float* __restrict__ h, const int* __restrict__ src,
    const int* __restrict__ dst, const float* __restrict__ ew,
    float* __restrict__ hi)
{
    int e = blockIdx.x * 8 + (threadIdx.x >> 5);
    if (e >= N_EDGES) return;
    int lane = threadIdx.x & 31;
    if (e + 4096 < N_EDGES) {           // stream the edge lists ahead of use
        __builtin_prefetch(src + e + 4096, 0, 1);
        __builtin_prefetch(dst + e + 4096, 0, 1);
        __builtin_prefetch(ew  + e + 4096, 0, 1);
    }
    int s = src[e], d = dst[e];
    float w = ew[e];
    float4 v = ((const float4*)(h + (size_t)s * NHIDDEN))[lane];
    float* o = hi + (size_t)d * NHIDDEN + lane * 4;
    atomicAdd(o + 0, w * v.x);
    atomicAdd(o + 1, w * v.y);
    atomicAdd(o + 2, w * v.z);
    atomicAdd(o + 3, w * v.w);
}

__global__ __launch_bounds__(256) void zero_kernel(float4* __restrict__ p, int n4)
{
    int i = blockIdx.x * blockDim.x + threadIdx.x;
    if (i < n4) p[i] = make_float4(0.f, 0.f, 0.f, 0.f);
}

// ---------------------------------------------------------------------------
// Layer GEMM (fused): S = 0.9*hi + 0.1*h0 built on the fly;
// h = relu(theta * (S @ W) + (1-theta) * S).   [50000 x 128] @ [128 x 128]
// ---------------------------------------------------------------------------
__global__ __launch_bounds__(256) void layer_gemm_kernel(
    const float* __restrict__ hi, const float* __restrict__ h0,
    const float* __restrict__ w, float* __restrict__ hout, float theta)
{
    __shared__ __align__(16) float lw[NHIDDEN * NHIDDEN];   // 64 KB
    const int tid  = threadIdx.x;
    const int wave = tid >> 5;
    const int lane = tid & 31;
    const int lh   = lane >> 4;
    const int ln   = lane & 15;
    const int rowBase = blockIdx.x * 128 + wave * 16;
    const int row  = rowBase + ln;
    const int rowc = row < N_NODES ? row : N_NODES - 1;
    const size_t abase = (size_t)rowc * NHIDDEN + 2 * lh;

    for (int i = tid; i < NHIDDEN * NHIDDEN; i += 256) {
        int k = i >> 7, n = i & 127;
        lw[bswz(k, n, NHIDDEN)] = w[i];
    }
    __syncthreads();

    v8f acc[8];
    #pragma unroll
    for (int t = 0; t < 8; ++t)
        #pragma unroll
        for (int i = 0; i < 8; ++i) acc[t][i] = 0.0f;

    v2f a;
    a.x = 0.9f * hi[abase]     + 0.1f * h0[abase];
    a.y = 0.9f * hi[abase + 1] + 0.1f * h0[abase + 1];
    #pragma unroll 4
    for (int g = 0; g < 32; ++g) {
        const v2f* brow = (const v2f*)(lw + (size_t)(g * 2 + lh) * NHIDDEN * 2);
        v2f b[8];
        #pragma unroll
        for (int t = 0; t < 8; ++t) b[t] = brow[t * 16 + ln];
        v2f an;
        if (g < 31) {
            size_t off = abase + 4 * g + 4;
            an.x = 0.9f * hi[off]     + 0.1f * h0[off];
            an.y = 0.9f * hi[off + 1] + 0.1f * h0[off + 1];
        }
        #pragma unroll
        for (int t = 0; t < 8; ++t) acc[t] = WMMA_F32(a, b[t], acc[t]);
        a = an;
    }
    float omt = 1.0f - theta;
    #pragma unroll
    for (int t = 0; t < 8; ++t) {
        #pragma unroll
        for (int r = 0; r < 8; ++r) {
            int orow = rowBase + r + 8 * lh;
            int ocol = t * 16 + ln;
            if (orow < N_NODES) {
                size_t idx = (size_t)orow * NHIDDEN + ocol;
                float s = 0.9f * hi[idx] + 0.1f * h0[idx];
                float v = theta * acc[t][r] + omt * s;
                hout[idx] = v > 0.0f ? v : 0.0f;
            }
        }
    }
}

// ---------------------------------------------------------------------------
// GEMM out: logits = h @ w2 + b2     [50000 x 128] @ [128 x 64]
// ---------------------------------------------------------------------------
__global__ __launch_bounds__(256) void gemm_out_kernel(
    const float* __restrict__ h, const float* __restrict__ w2,
    const float* __restrict__ b2, float* __restrict__ logits)
{
    __shared__ __align__(16) float lw[NHIDDEN * NCLASS];    // 32 KB
    const int tid  = threadIdx.x;
    const int wave = tid >> 5;
    const int lane = tid & 31;
    const int lh   = lane >> 4;
    const int ln   = lane & 15;
    const int rowBase = blockIdx.x * 128 + wave * 16;
    const int row  = rowBase + ln;
    const int rowc = row < N_NODES ? row : N_NODES - 1;
    const float* arow = h + (size_t)rowc * NHIDDEN + 2 * lh;

    for (int i = tid; i < NHIDDEN * NCLASS; i += 256) {
        int k = i >> 6, n = i & 63;
        lw[bswz(k, n, NCLASS)] = w2[i];
    }
    __syncthreads();

    v8f acc[4];
    #pragma unroll
    for (int t = 0; t < 4; ++t)
        #pragma unroll
        for (int i = 0; i < 8; ++i) acc[t][i] = 0.0f;

    v2f a; a.x = arow[0]; a.y = arow[1];
    #pragma unroll 4
    for (int g = 0; g < 32; ++g) {
        const v2f* brow = (const v2f*)(lw + (size_t)(g * 2 + lh) * NCLASS * 2);
        v2f b[4];
        #pragma unroll
        for (int t = 0; t < 4; ++t) b[t] = brow[t * 16 + ln];
        v2f an;
        if (g < 31) { an.x = arow[4 * g + 4]; an.y = arow[4 * g + 5]; }
        #pragma unroll
        for (int t = 0; t < 4; ++t) acc[t] = WMMA_F32(a, b[t], acc[t]);
        a = an;
    }
    #pragma unroll
    for (int t = 0; t < 4; ++t) {
        #pragma unroll
        for (int r = 0; r < 8; ++r) {
            int orow = rowBase + r + 8 * lh;
            int ocol = t * 16 + ln;
            if (orow < N_NODES)
                logits[(size_t)orow * NCLASS + ocol] = acc[t][r] + b2[ocol];
        }
    }
}

// ---------------------------------------------------------------------------
// log_softmax over 64 classes: one wave per row, 2 values/lane, shfl reduce.
// ---------------------------------------------------------------------------
__global__ __launch_bounds__(256) void logsoftmax_kernel(
    const float* __restrict__ logits, float* __restrict__ out)
{
    int row = blockIdx.x * 8 + (threadIdx.x >> 5);
    if (row >= N_NODES) return;
    int lane = threadIdx.x & 31;
    float2 v = ((const float2*)(logits + (size_t)row * NCLASS))[lane];
    float m = fmaxf(v.x, v.y);
    #pragma unroll
    for (int o = 16; o > 0; o >>= 1) m = fmaxf(m, __shfl_xor(m, o, 32));
    float s = expf(v.x - m) + expf(v.y - m);
    #pragma unroll
    for (int o = 16; o > 0; o >>= 1) s += __shfl_xor(s, o, 32);
    float lse = m + logf(s);
    float2 r; r.x = v.x - lse; r.y = v.y - lse;
    ((float2*)(out + (size_t)row * NCLASS))[lane] = r;
}

// ---------------------------------------------------------------------------
extern "C" void kernel_launch(void* const* d_in, const int* in_sizes, int n_in,
                              void* d_out, int out_size, void* d_ws, size_t ws_size,
                              hipStream_t stream)
{
    (void)in_sizes; (void)n_in; (void)out_size; (void)ws_size;
    const float* x     = (const float*)d_in[0];
    const int*   esrc  = (const int*)  d_in[1];
    const int*   edst  = (const int*)  d_in[2];
    const float* ew    = (const float*)d_in[3];
    const float* w1    = (const float*)d_in[4];
    const float* b1    = (const float*)d_in[5];
    const float* convw = (const float*)d_in[6];
    const float* w2    = (const float*)d_in[7];
    const float* b2    = (const float*)d_in[8];
    float* out = (float*)d_out;

    const size_t HB = (size_t)N_NODES * NHIDDEN;   // 6.4M floats
    float* h0     = (float*)d_ws;
    float* h      = h0 + HB;
    float* hi     = h  + HB;
    float* logits = hi + HB;

    dim3 blk(256);
    const int mblocks = (N_NODES + 127) / 128;
    const int n4      = (int)(HB / 4);
    const int zblocks = (n4 + 255) / 256;
    const int eblocks = (N_EDGES + 7) / 8;

    gemm_in_kernel<<<mblocks, blk, 0, stream>>>(x, w1, b1, h0);

    for (int l = 0; l < NLAYERS; ++l) {
        float theta = logf(LAMDA / (float)(l + 1) + 1.0f);
        zero_kernel<<<zblocks, blk, 0, stream>>>((float4*)hi, n4);
        const float* hin = (l == 0) ? h0 : h;
        spmm_kernel<<<eblocks, blk, 0, stream>>>(hin, esrc, edst, ew, hi);
        layer_gemm_kernel<<<mblocks, blk, 0, stream>>>(
            hi, h0, convw + (size_t)l * NHIDDEN * NHIDDEN, h, theta);
    }

    gemm_out_kernel<<<mblocks, blk, 0, stream>>>(h, w2, b2, logits);
    logsoftmax_kernel<<<(N_NODES + 7) / 8, blk, 0, stream>>>(logits, out);
}